// CrossEntropyLoss_surv_75325136437599
// MI455X (gfx1250) — compile-verified
//
#include <hip/hip_runtime.h>
#include <hip/hip_bf16.h>

typedef __attribute__((ext_vector_type(2))) float v2f;
typedef __attribute__((ext_vector_type(8))) float v8f;

#define KBINS 64
#define ROWS_PER_WAVE 16
#define WAVES_PER_BLOCK 8
#define ROWS_PER_BLOCK (ROWS_PER_WAVE * WAVES_PER_BLOCK)
#define LDS_ROW 68   // 64 data floats + 4 pad -> conflict-free ds_load_b64 pattern

__device__ __forceinline__ int iminf(int a, int b) { return a < b ? a : b; }

// S0 = exp(-4) in fp32, matching the reference's jnp.exp(-4.0)
#define S0_CONST 0.01831563889f

// Branchless 64-bit label mask: bit j == count[row][j]
__device__ __forceinline__ unsigned long long label_mask(int b, int valid, int ev) {
    const int bs = b & 63;                       // avoid UB shifts when invalid
    const unsigned long long onehot = 1ull << bs;
    const unsigned long long tail   = (~0ull) << bs;
    unsigned long long m = (ev != 0) ? onehot : tail;   // value select -> v_cndmask
    return valid ? m : 0ull;
}

__global__ __launch_bounds__(256) void surv_loss_kernel(
    const float* __restrict__ inp, const float* __restrict__ y,
    const int* __restrict__ e, int* __restrict__ outCount,
    float* __restrict__ blockPart, int B)
{
    __shared__ float tile[WAVES_PER_BLOCK * ROWS_PER_WAVE * LDS_ROW];
    __shared__ float wpart[WAVES_PER_BLOCK * 2];

    const int tid  = threadIdx.x;
    const int wave = tid >> 5;
    const int lane = tid & 31;
    const int rowBase = blockIdx.x * ROWS_PER_BLOCK + wave * ROWS_PER_WAVE;
    const int Bm1 = B - 1;
    const int wbase = wave * ROWS_PER_WAVE * LDS_ROW;

    // ---- Stage 16 rows (16x64 f32 = 4KB) into padded LDS, coalesced b128 ----
#pragma unroll
    for (int t = 0; t < 8; ++t) {
        int g   = t * 32 + lane;      // 0..255 float4 chunks
        int row = g >> 4;             // 0..15
        int c4  = g & 15;             // 0..15
        long rg = iminf(rowBase + row, Bm1);
        const float4 v = *(const float4*)(inp + rg * (long)KBINS + c4 * 4);
        *(float4*)(&tile[wbase + row * LDS_ROW + c4 * 4]) = v;
    }
    __syncthreads();

    // ---- Per-lane label metadata for the WMMA A-matrix layout ----
    // A (16x4 f32): lanes 0-15 hold M=lane, K=0..1 ; lanes 16-31 hold M=lane-16, K=2..3
    const int mrow = lane & 15;
    const int sub  = (lane >> 4) * 2;
    float yv; int ev;
    {
        long rg = iminf(rowBase + mrow, Bm1);
        yv = y[rg];
        ev = e[rg];
    }
    const int b     = (int)floorf(yv);
    const int valid = (b >= 0) & (b < KBINS);
    const unsigned long long mask = label_mask(b, valid, ev);
    const unsigned mlo = (unsigned)mask;
    const unsigned mhi = (unsigned)(mask >> 32);
    // min(count)==1 only when censored, valid, b==0 (tail covers every bin)
    const int minc1 = valid & (ev == 0) & (b == 0);

    v8f accA = {};   // correct  = sum exp(-x)*count
    v8f accB = {};   // false_ev = sum exp(+x)*(1-count)
    v8f accC = {};   // false_cs = sum exp(-x)*(1-count)
    v2f ones = {1.0f, 1.0f};

#pragma unroll
    for (int c = 0; c < 16; ++c) {
        const int cbase = c * 4;                      // compile-time constant
        const unsigned w = (cbase < 32) ? mlo : mhi;  // folds at compile time
        const int sh0 = (cbase & 31) + sub;
        const int sh1 = sh0 + 1;

        v2f x = *(const v2f*)(&tile[wbase + mrow * LDS_ROW + cbase + sub]);

        const int bit0 = (int)((w >> sh0) & 1u);
        const int bit1 = (int)((w >> sh1) & 1u);
        const float c0 = bit0 ? 1.0f : 0.0f;
        const float c1 = bit1 ? 1.0f : 0.0f;
        // margin where count == min(count); exact match with reference rounding
        const float l0 = (x[0] + ((bit0 == minc1) ? 0.1f : 0.0f)) * 2.0f;
        const float l1 = (x[1] + ((bit1 == minc1) ? 0.1f : 0.0f)) * 2.0f;

        // |l| <= ~11.3: no overflow region, fast exp is safe (~1e-6 rel err)
        const float pen0 = __expf(l0),  pen1 = __expf(l1);
        const float pe0  = __expf(-l0), pe1  = __expf(-l1);

        const float a0 = pe0 * c0,   a1 = pe1 * c1;    // pe*count
        const float t0 = pen0 * c0,  t1 = pen1 * c1;   // pen*count
        v2f aA = { a0,         a1 };
        v2f aB = { pen0 - t0,  pen1 - t1 };            // pen*(1-count)
        v2f aC = { pe0 - a0,   pe1 - a1 };             // pe*(1-count)

        // D = A * ones(4x16) + C  ->  fp32 row sums accumulated in-register
        accA = __builtin_amdgcn_wmma_f32_16x16x4_f32(false, aA, false, ones, (short)0, accA, false, false);
        accB = __builtin_amdgcn_wmma_f32_16x16x4_f32(false, aB, false, ones, (short)0, accB, false, false);
        accC = __builtin_amdgcn_wmma_f32_16x16x4_f32(false, aC, false, ones, (short)0, accC, false, false);
    }

    // ---- Per-sample loss: lanes 0-15 hold rows 0..7 in accX[0..7], lanes 16-31 rows 8..15 ----
    float part = 0.0f;
    const int mb = (lane >> 4) * 8;
#pragma unroll
    for (int i = 0; i < 8; ++i) {
        const int m  = mb + i;
        const int rg = rowBase + m;
        const int rc = iminf(rg, Bm1);
        const float yv2 = y[rc];
        const int   ev2 = e[rc];
        const int   b2  = (int)floorf(yv2);
        const int   valid2 = (b2 >= 0) & (b2 < KBINS);
        const float corr = accA[i];
        const float fev  = accB[i];
        const float fcs  = accC[i];
        // sum(count)==1  <=>  valid && (event || b == K-1)
        const int isev = valid2 & ((ev2 != 0) | (b2 == KBINS - 1));
        const float loss_ev = logf(corr + S0_CONST) + logf(fev + S0_CONST);
        const float loss_cs = logf(1.0f + corr * fcs + fcs / S0_CONST);
        const float loss = isev ? loss_ev : loss_cs;
        part += (rg < B) ? loss : 0.0f;
    }
    if (lane == 0)  wpart[wave * 2 + 0] = part;  // uniform across lanes 0..15
    if (lane == 16) wpart[wave * 2 + 1] = part;  // uniform across lanes 16..31

    // ---- count output: regenerate from masks; coalesced b32 stores (region is 4B-aligned) ----
    for (int row = 0; row < 16; ++row) {
        const int rg = rowBase + row;
        const int rc = iminf(rg, Bm1);
        const float yv3 = y[rc];
        const int   ev3 = e[rc];
        const int   b3  = (int)floorf(yv3);
        const int   valid3 = (b3 >= 0) & (b3 < KBINS);
        const unsigned long long m3 = label_mask(b3, valid3, ev3);
        const unsigned w0 = (unsigned)m3;
        const unsigned w1 = (unsigned)(m3 >> 32);
        if (rg < B) {
            outCount[(long)rg * KBINS + lane]      = (int)((w0 >> lane) & 1u);
            outCount[(long)rg * KBINS + 32 + lane] = (int)((w1 >> lane) & 1u);
        }
    }

    __syncthreads();
    if (tid == 0) {  // fixed-order block reduce -> deterministic
        float s = 0.0f;
#pragma unroll
        for (int i = 0; i < WAVES_PER_BLOCK * 2; ++i) s += wpart[i];
        blockPart[blockIdx.x] = s;
    }
}

__global__ __launch_bounds__(256) void surv_reduce_kernel(
    const float* __restrict__ part, int n, float* __restrict__ out, int B)
{
    __shared__ double sm[256];
    double a = 0.0;
    for (int i = threadIdx.x; i < n; i += 256) a += (double)part[i];  // fixed order
    sm[threadIdx.x] = a;
    __syncthreads();
    for (int off = 128; off > 0; off >>= 1) {
        if (threadIdx.x < off) sm[threadIdx.x] += sm[threadIdx.x + off];
        __syncthreads();
    }
    if (threadIdx.x == 0) out[0] = (float)(sm[0] / (double)B);
}

extern "C" void kernel_launch(void* const* d_in, const int* in_sizes, int n_in,
                              void* d_out, int out_size, void* d_ws, size_t ws_size,
                              hipStream_t stream) {
    const float* inp = (const float*)d_in[0];   // [B, 64] f32
    const float* y   = (const float*)d_in[1];   // [B] f32
    const int*   e   = (const int*)d_in[2];     // [B] i32
    const int B = in_sizes[1];

    float* outLoss  = (float*)d_out;            // element 0: scalar loss
    int*   outCount = ((int*)d_out) + 1;        // elements 1..B*K: int32 count
    float* blockPart = (float*)d_ws;            // nBlocks floats of scratch

    const int nBlocks = (B + ROWS_PER_BLOCK - 1) / ROWS_PER_BLOCK;  // 4096 for B=524288
    surv_loss_kernel<<<nBlocks, 256, 0, stream>>>(inp, y, e, outCount, blockPart, B);
    surv_reduce_kernel<<<1, 256, 0, stream>>>(blockPart, nBlocks, outLoss, B);
}